// Gridding_5669356836198
// MI455X (gfx1250) — compile-verified
//
#include <hip/hip_runtime.h>

#define THREADS 256
#define MAGIC_SHIFT 45

struct GridParams {
    int s;                    // half-scale
    int G;                    // vertices per axis
    int G3;                   // cells per batch
    int pad;
    unsigned long long m;     // magic multiplier: batch = (gid * m) >> MAGIC_SHIFT
};

// ---------------------------------------------------------------------------
// Kernel A: trilinear scatter. One point per lane (wave32): one b96 coalesced
// load, ~20 VALU ops, 8 no-return global_atomic_add_f32 into the per-batch
// grid (8 MB/batch -> resident in the 192 MB L2; atomics RMW at L2).
// Batch index via branchless magic multiply -> no division code in binary.
// ---------------------------------------------------------------------------
__global__ __launch_bounds__(THREADS) void grid_scatter_kernel(
        const float*      __restrict__ pt,   // [B*N*3]
        const GridParams* __restrict__ P,    // precomputed uniforms (d_ws)
        float*            __restrict__ out,  // [B*G^3]
        int total)                           // B*N
{
    int gid = blockIdx.x * THREADS + threadIdx.x;
    if (gid >= total) return;

    int s  = P->s;
    int G  = P->G;
    int G3 = P->G3;
    unsigned long long m = P->m;

    float fs = (float)s;
    const float* pp = pt + (size_t)gid * 3;   // 12B/lane contiguous -> b96
    float x = pp[0] * fs;
    float y = pp[1] * fs;
    float z = pp[2] * fs;

    // Reference zero-weights points whose scaled coords sum to exactly 0
    // (padding). Keep the reference's rounding: plain adds of the already-
    // rounded products (no FMA contraction). Weights >= 0, so skip == +0.0.
    float sum;
    {
#pragma clang fp contract(off)
        sum = (x + y) + z;
    }
    if (sum == 0.0f) return;

    float lx = floorf(x), ly = floorf(y), lz = floorf(z);
    float ax = x - lx,  ay = y - ly,  az = z - lz;      // frac  (upper corner)
    float bx = 1.f - ax, by = 1.f - ay, bz = 1.f - az;  // 1-frac (lower corner)

    int ix = (int)lx + s;   // in [0,G) by construction of the inputs
    int iy = (int)ly + s;
    int iz = (int)lz + s;

    // Exact floor(gid/N) for 2^12 <= N <= 2^14, gid < 2^31 (pow-2 N reduces to shift).
    unsigned batch = (unsigned)(((unsigned long long)(unsigned)gid * m) >> MAGIC_SHIFT);

    size_t cell = ((size_t)ix * G + iy) * (size_t)G + iz;
    float* o = out + (size_t)batch * (size_t)G3 + cell;
    size_t dG  = (size_t)G;       // +1 in y
    size_t dGG = (size_t)G * G;   // +1 in x

    // _OFF order: c=(i,j,k) over x,y,z; flat = ((ix+i)*G + iy+j)*G + iz+k.
    // No-return f32 atomics -> hardware global_atomic_add_f32 (STOREcnt).
    unsafeAtomicAdd(o,                (bx * by) * bz);   // 000
    unsafeAtomicAdd(o + 1,            (bx * by) * az);   // 001
    unsafeAtomicAdd(o + dG,           (bx * ay) * bz);   // 010
    unsafeAtomicAdd(o + dG + 1,       (bx * ay) * az);   // 011
    unsafeAtomicAdd(o + dGG,          (ax * by) * bz);   // 100
    unsafeAtomicAdd(o + dGG + 1,      (ax * by) * az);   // 101
    unsafeAtomicAdd(o + dGG + dG,     (ax * ay) * bz);   // 110
    unsafeAtomicAdd(o + dGG + dG + 1, (ax * ay) * az);   // 111
}

// ---------------------------------------------------------------------------
// Kernel B: derive uniform params once (scale lives in device memory, so the
// host can never see it under graph capture). Single lane; all integer
// division lives here instead of in 65k wave prologues.
// ---------------------------------------------------------------------------
__global__ void grid_setup_kernel(const int* __restrict__ scale_p,
                                  int total, int out_size,
                                  GridParams* __restrict__ P)
{
    int sc = *scale_p;          // 128
    int s  = sc >> 1;           // 64
    int G  = s << 1;            // 128
    int G3 = G * G * G;         // 2^21 cells per batch
    int B  = out_size / G3;     // 32
    int N  = total / B;         // 65536 points per batch
    P->s  = s;
    P->G  = G;
    P->G3 = G3;
    P->m  = ((1ULL << MAGIC_SHIFT) + (unsigned)N - 1) / (unsigned)N;
}

// ---------------------------------------------------------------------------
// Kernel C: zero the 256 MB output. Grid-stride float4 stores (regular
// temporal hint so zeroed lines stay in L2 for the atomics that follow).
// ---------------------------------------------------------------------------
__global__ __launch_bounds__(THREADS) void grid_zero_kernel(float* __restrict__ out, int n) {
    int n4 = n >> 2;
    float4* out4 = (float4*)out;
    float4 z4 = make_float4(0.f, 0.f, 0.f, 0.f);
    int stride = gridDim.x * THREADS;
    for (int k = blockIdx.x * THREADS + threadIdx.x; k < n4; k += stride)
        out4[k] = z4;
    if (blockIdx.x == 0 && threadIdx.x == 0)            // tail (none when n%4==0)
        for (int k = n4 << 2; k < n; ++k) out[k] = 0.f;
}

// ---------------------------------------------------------------------------
extern "C" void kernel_launch(void* const* d_in, const int* in_sizes, int n_in,
                              void* d_out, int out_size, void* d_ws, size_t ws_size,
                              hipStream_t stream) {
    const float* pt    = (const float*)d_in[0];   // [B,N,3] float32
    const int*   scale = (const int*)  d_in[1];   // scalar int (device)
    float*       out   = (float*)d_out;           // [B, G^3] float32
    GridParams*  P     = (GridParams*)d_ws;       // scratch params

    int total = in_sizes[0] / 3;                  // B*N points

    grid_setup_kernel<<<1, 1, 0, stream>>>(scale, total, out_size, P);

    int n4 = out_size >> 2;
    int zblocks = (n4 + THREADS - 1) / THREADS;
    if (zblocks > 16384) zblocks = 16384;         // ~4 float4 stores per thread
    grid_zero_kernel<<<zblocks, THREADS, 0, stream>>>(out, out_size);

    int sblocks = (total + THREADS - 1) / THREADS;
    grid_scatter_kernel<<<sblocks, THREADS, 0, stream>>>(pt, P, out, total);
}